// GSCLoss_75608604279422
// MI455X (gfx1250) — compile-verified
//
#include <hip/hip_runtime.h>
#include <cstdint>
#include <cstddef>

#define TILE 256
#define KEYS 8
#define FLOATS_PER_B 24   // 8 keypoints * 3 comps
#define BYTES_PER_B  96
#define EPSF 1e-6f

// ---------------- CDNA5 async global->LDS staging helpers ----------------
#if defined(__HIP_DEVICE_COMPILE__) && __has_builtin(__builtin_amdgcn_global_load_async_to_lds_b128)
#define USE_ASYNC_LDS 1
#else
#define USE_ASYNC_LDS 0
#endif

#if USE_ASYNC_LDS
// Builtin signature (from hipcc diagnostic): first param is
// 'int __attribute__((vector_size(16))) __device__ *' i.e. AS(1) v4i32*.
typedef int v4i32 __attribute__((vector_size(16)));
typedef __attribute__((address_space(1))) v4i32 g_v4i32;
typedef __attribute__((address_space(3))) v4i32 l_v4i32;

__device__ __forceinline__ void async_copy_b128(const void* g, void* l) {
  // generic->AS1: flat global address is the global address.
  // generic->AS3: low 32 bits of a flat LDS address are the LDS offset.
  __builtin_amdgcn_global_load_async_to_lds_b128(
      (g_v4i32*)(uintptr_t)g,
      (l_v4i32*)(uint32_t)(uintptr_t)l,
      /*offset=*/0, /*cpol=*/0);
}

__device__ __forceinline__ void wait_async_done() {
#if __has_builtin(__builtin_amdgcn_s_wait_asynccnt)
  __builtin_amdgcn_s_wait_asynccnt(0);
#else
  asm volatile("s_wait_asynccnt 0" ::: "memory");
#endif
}
#endif

__device__ __forceinline__ float n2(float x, float y) {
  return sqrtf(x * x + y * y);
}

// ---------------- Pass 1: per-block partial (sum_L, count_valid) ----------------
__global__ __launch_bounds__(TILE) void gsc_partial(const float* __restrict__ pred,
                                                    const float* __restrict__ gt,
                                                    float* __restrict__ partial,
                                                    int B) {
  __shared__ __align__(16) float s_pred[TILE * FLOATS_PER_B];  // 24 KB
  __shared__ __align__(16) float s_gt  [TILE * FLOATS_PER_B];  // 24 KB
  __shared__ float s_sum[TILE];
  __shared__ float s_cnt[TILE];

  const int tid        = threadIdx.x;
  const int blockStart = blockIdx.x * TILE;
  const int nb         = (B - blockStart < TILE) ? (B - blockStart) : TILE;
  const int nchunks    = (nb * BYTES_PER_B) >> 4;  // 16-byte chunks (96 % 16 == 0)

  const char* gp = (const char*)(pred + (size_t)blockStart * FLOATS_PER_B);
  const char* gg = (const char*)(gt   + (size_t)blockStart * FLOATS_PER_B);

#if USE_ASYNC_LDS
  for (int c = tid; c < nchunks; c += TILE)
    async_copy_b128(gp + ((size_t)c << 4), (char*)s_pred + ((size_t)c << 4));
  for (int c = tid; c < nchunks; c += TILE)
    async_copy_b128(gg + ((size_t)c << 4), (char*)s_gt + ((size_t)c << 4));
  wait_async_done();            // this wave's async copies are in LDS
#else
  for (int c = tid; c < nchunks; c += TILE) {
    ((float4*)s_pred)[c] = ((const float4*)gp)[c];
    ((float4*)s_gt)[c]   = ((const float4*)gg)[c];
  }
#endif
  __syncthreads();              // every wave waited on its own copies -> tile complete

  float contrib = 0.0f;
  float cnt     = 0.0f;

  if (tid < nb) {
    const float* P = s_pred + tid * FLOATS_PER_B;
    const float* G = s_gt   + tid * FLOATS_PER_B;

    // Stable selection: indices of the first 4 visible keypoints.
    // If fewer than 4 are visible the batch is invalid and contributes 0,
    // so the fill values are irrelevant (reference masks L with where()).
    int n = 0;
    int i0 = 0, i1 = 0, i2 = 0, i3 = 0;
#pragma unroll
    for (int k = 0; k < KEYS; ++k) {
      const bool v = (P[3 * k + 2] > 0.5f) && (G[3 * k + 2] > 0.5f);
      if (v) {
        if (n == 0) i0 = k;
        else if (n == 1) i1 = k;
        else if (n == 2) i2 = k;
        else if (n == 3) i3 = k;
        ++n;
      }
    }

    if (n >= 4) {
      // Dynamic-index gathers straight from LDS (ds_load_b32 w/ VGPR addr).
      const float p1x = P[3 * i0], p1y = P[3 * i0 + 1];
      const float p2x = P[3 * i1], p2y = P[3 * i1 + 1];
      const float p3x = P[3 * i2], p3y = P[3 * i2 + 1];
      const float p4x = P[3 * i3], p4y = P[3 * i3 + 1];
      const float g1x = G[3 * i0], g1y = G[3 * i0 + 1];
      const float g2x = G[3 * i1], g2y = G[3 * i1 + 1];
      const float g3x = G[3 * i2], g3y = G[3 * i2 + 1];
      const float g4x = G[3 * i3], g4y = G[3 * i3 + 1];

      // L_shape: cross-ratio of diagonals
      const float pred_cr = n2(p3x - p1x, p3y - p1y) / (n2(p4x - p2x, p4y - p2y) + EPSF);
      const float gt_cr   = n2(g3x - g1x, g3y - g1y) / (n2(g4x - g2x, g4y - g2y) + EPSF);
      const float L_shape = fabsf(pred_cr - gt_cr);

      // L_edge
      const float v12x = p2x - p1x, v12y = p2y - p1y;
      const float v23x = p3x - p2x, v23y = p3y - p2y;
      const float v34x = p4x - p3x, v34y = p4y - p3y;
      const float v41x = p1x - p4x, v41y = p1y - p4y;
      const float l12 = n2(v12x, v12y), l23 = n2(v23x, v23y);
      const float l34 = n2(v34x, v34y), l41 = n2(v41x, v41y);
      const float par1 = fabsf(l12 - l34) / (l12 + l34 + EPSF);
      const float par2 = fabsf(l23 - l41) / (l23 + l41 + EPSF);
      const float dot1 = fabsf((v12x * v41x + v12y * v41y) / (l12 * l41 + EPSF));
      const float dot2 = fabsf((v12x * v23x + v12y * v23y) / (l12 * l23 + EPSF));
      const float dot3 = fabsf((v23x * v34x + v23y * v34y) / (l23 * l34 + EPSF));
      const float dot4 = fabsf((v34x * v41x + v34y * v41y) / (l34 * l41 + EPSF));
      const float L_edge = 0.5f * (par1 + par2) + 0.25f * (dot1 + dot2 + dot3 + dot4);

      // L_pos
      const float dist = 0.25f * (n2(p1x - g1x, p1y - g1y) + n2(p2x - g2x, p2y - g2y) +
                                  n2(p3x - g3x, p3y - g3y) + n2(p4x - g4x, p4y - g4y));

      const float pa = 0.5f * fabsf((p2x - p1x) * (p3y - p1y) - (p2y - p1y) * (p3x - p1x)) +
                       0.5f * fabsf((p3x - p1x) * (p4y - p1y) - (p3y - p1y) * (p4x - p1x));
      const float ga = 0.5f * fabsf((g2x - g1x) * (g3y - g1y) - (g2y - g1y) * (g3x - g1x)) +
                       0.5f * fabsf((g3x - g1x) * (g4y - g1y) - (g3y - g1y) * (g4x - g1x));
      const float area_ratio = fabsf(pa - ga) / (ga + EPSF);

      const float pmx = 0.25f * (p1x + p2x + p3x + p4x), pmy = 0.25f * (p1y + p2y + p3y + p4y);
      const float gmx = 0.25f * (g1x + g2x + g3x + g4x), gmy = 0.25f * (g1y + g2y + g3y + g4y);
      const float rel = 0.25f * (n2((p1x - pmx) - (g1x - gmx), (p1y - pmy) - (g1y - gmy)) +
                                 n2((p2x - pmx) - (g2x - gmx), (p2y - pmy) - (g2y - gmy)) +
                                 n2((p3x - pmx) - (g3x - gmx), (p3y - pmy) - (g3y - gmy)) +
                                 n2((p4x - pmx) - (g4x - gmx), (p4y - pmy) - (g4y - gmy)));

      const float L_pos = 0.4f * dist + 0.3f * area_ratio + 0.3f * rel;
      contrib = 0.4f * L_shape + 0.3f * L_edge + 0.3f * L_pos;
      cnt = 1.0f;
    }
  }

  // Deterministic block tree-reduction of (sum, count)
  s_sum[tid] = contrib;
  s_cnt[tid] = cnt;
  __syncthreads();
#pragma unroll
  for (int s = TILE / 2; s > 0; s >>= 1) {
    if (tid < s) {
      s_sum[tid] += s_sum[tid + s];
      s_cnt[tid] += s_cnt[tid + s];
    }
    __syncthreads();
  }
  if (tid == 0) {
    partial[2 * blockIdx.x]     = s_sum[0];
    partial[2 * blockIdx.x + 1] = s_cnt[0];
  }
}

// ---------------- Pass 2: reduce partials, write scalar ----------------
__global__ __launch_bounds__(256) void gsc_final(const float* __restrict__ partial,
                                                 int nPart,
                                                 float* __restrict__ out) {
  __shared__ float rs[256];
  __shared__ float rc[256];
  float s = 0.0f, c = 0.0f;
  for (int i = threadIdx.x; i < nPart; i += 256) {
    s += partial[2 * i];
    c += partial[2 * i + 1];
  }
  rs[threadIdx.x] = s;
  rc[threadIdx.x] = c;
  __syncthreads();
#pragma unroll
  for (int st = 128; st > 0; st >>= 1) {
    if (threadIdx.x < st) {
      rs[threadIdx.x] += rs[threadIdx.x + st];
      rc[threadIdx.x] += rc[threadIdx.x + st];
    }
    __syncthreads();
  }
  if (threadIdx.x == 0) {
    out[0] = (rc[0] > 0.0f) ? (rs[0] / rc[0]) : 0.0f;
  }
}

// ---------------- Host entry ----------------
extern "C" void kernel_launch(void* const* d_in, const int* in_sizes, int n_in,
                              void* d_out, int out_size, void* d_ws, size_t ws_size,
                              hipStream_t stream) {
  const float* pred = (const float*)d_in[0];
  const float* gt   = (const float*)d_in[1];
  const int B       = in_sizes[0] / FLOATS_PER_B;  // 524288
  const int nBlocks = (B + TILE - 1) / TILE;       // 2048 -> 16 KB of partials

  float* partial = (float*)d_ws;

  gsc_partial<<<nBlocks, TILE, 0, stream>>>(pred, gt, partial, B);
  gsc_final<<<1, 256, 0, stream>>>(partial, nBlocks, (float*)d_out);
}